// AWingLoss_22883585753113
// MI455X (gfx1250) — compile-verified
//
#include <hip/hip_runtime.h>
#include <math.h>

// ---------------------------------------------------------------------------
// AWing loss, fused dilation+mask+mean, MI455X (gfx1250, wave32).
// Bandwidth-bound target: ~285 MB in / 4 B out -> ~12us floor at 23.3 TB/s.
// - target tile (+1-row halo) staged into LDS via CDNA5 async copies
//   (global_load_async_to_lds_b128 + s_wait_asynccnt)   [confirmed in disasm]
// - 4-pixel quads: b128 global loads, separable 3x3 max from LDS
// - transcendentals on raw v_exp_f32/v_log_f32 (5 trans + ~30 VALU per pixel)
//   to stay under the HBM roofline instead of the device-lib log1pf path.
// ---------------------------------------------------------------------------

#define THREADS 256
#define ROWS    32                 // rows of one plane per block
#define HALO    (ROWS + 2)
#define WD      128
#define HD      128
#define PLANES  (32 * 68)
#define BLKS_PER_PLANE (HD / ROWS) // 4
#define NBLOCKS (PLANES * BLKS_PER_PLANE)  // 8704
#define NTOT    ((long long)PLANES * HD * WD) // 35,651,584

#if defined(__HIP_DEVICE_COMPILE__) && \
    __has_builtin(__builtin_amdgcn_global_load_async_to_lds_b128) && \
    __has_builtin(__builtin_amdgcn_s_wait_asynccnt)
#define USE_ASYNC_LDS 1
typedef int v4i __attribute__((vector_size(16)));
typedef __attribute__((address_space(1))) v4i GV4;
typedef __attribute__((address_space(3))) v4i LV4;
#else
#define USE_ASYNC_LDS 0
#endif

#if defined(__HIP_DEVICE_COMPILE__) && __has_builtin(__builtin_amdgcn_exp2f)
#define EXP2F(x) __builtin_amdgcn_exp2f(x)     // v_exp_f32: exp2(-inf)=0 in HW
#else
#define EXP2F(x) exp2f(x)
#endif
#if defined(__HIP_DEVICE_COMPILE__) && __has_builtin(__builtin_amdgcn_logf)
#define LOG2F(x) __builtin_amdgcn_logf(x)      // v_log_f32: log2(0)=-inf in HW
#else
#define LOG2F(x) log2f(x)
#endif

__device__ __forceinline__ float max3(float a, float b, float c) {
    return fmaxf(fmaxf(a, b), c);
}

// log1p for x in [0, 1.5): log2(1+x)*ln2. Rel err ~1e-6; for x<<1 the absolute
// error is <~1e-7 which is negligible in a mean over 35.6M elements.
__device__ __forceinline__ float log1p_fast(float x) {
    return LOG2F(1.0f + x) * 0.6931471805599453f;
}

__global__ __launch_bounds__(THREADS)
void awing_main(const float* __restrict__ pred,
                const float* __restrict__ tgt,
                float* __restrict__ partial)
{
    __shared__ float tile[HALO * WD];            // 17.4 KB
    __shared__ float wsum[THREADS / 32];

    const int tid   = threadIdx.x;
    const int blk   = blockIdx.x;
    const int plane = blk / BLKS_PER_PLANE;
    const int rblk  = blk % BLKS_PER_PLANE;
    const int g0    = rblk * ROWS;               // first output row of slab

    const float* tp = tgt  + (size_t)plane * (HD * WD);
    const float* pp = pred + (size_t)plane * (HD * WD);

    // ---- stage target rows [g0-1, g0+ROWS] (row-clamped) into LDS ----------
    // Clamping duplicates an already-in-window value, so the 3x3 max equals
    // the reference's -inf-padded reduce_window / reflect dilation.
    const int n4 = HALO * (WD / 4);              // 1088 float4 chunks
    for (int i = tid; i < n4; i += THREADS) {
        const int tr  = i >> 5;                  // 0..HALO-1
        const int tc4 = (i & 31) << 2;           // col in floats, /4 aligned
        int gr = g0 - 1 + tr;
        gr = gr < 0 ? 0 : (gr > HD - 1 ? HD - 1 : gr);
#if USE_ASYNC_LDS
        __builtin_amdgcn_global_load_async_to_lds_b128(
            (GV4*)(tp + gr * WD + tc4),
            (LV4*)(&tile[tr * WD + tc4]),
            0, 0);
#else
        const float4 v = *(const float4*)(tp + gr * WD + tc4);
        *(float4*)(&tile[tr * WD + tc4]) = v;
#endif
    }
#if USE_ASYNC_LDS
    __builtin_amdgcn_s_wait_asynccnt(0);
#endif
    __syncthreads();

    // ---- fused dilation + AWing loss + mask, 4-pixel quads -----------------
    // 32 quad-lanes cover one 128-wide row; 8 row-groups; 4 iterations = 32 rows.
    const int col4 = (tid & 31) << 2;            // 0,4,...,124
    const int rgrp = tid >> 5;                   // 0..7
    const int cmL  = (col4 == 0)      ? 0      : col4 - 1;
    const int cpR  = (col4 == WD - 4) ? WD - 1 : col4 + 4;

    float lsum = 0.0f;
    #pragma unroll
    for (int it = 0; it < ROWS / 8; ++it) {
        const int lr = (it << 3) + rgrp;         // local row 0..ROWS-1
        const float* r0 = &tile[lr * WD];        // row above (tile row lr)
        const float* r1 = r0 + WD;               // center row
        const float* r2 = r1 + WD;               // row below

        // separable 3x3 max: vertical maxes of 6 columns, then horizontal 3-max
        const float4 a0 = *(const float4*)(r0 + col4);
        const float4 a1 = *(const float4*)(r1 + col4);
        const float4 a2 = *(const float4*)(r2 + col4);
        const float vL = max3(r0[cmL], r1[cmL], r2[cmL]);
        const float vR = max3(r0[cpR], r1[cpR], r2[cpR]);
        const float v0 = max3(a0.x, a1.x, a2.x);
        const float v1 = max3(a0.y, a1.y, a2.y);
        const float v2 = max3(a0.z, a1.z, a2.z);
        const float v3 = max3(a0.w, a1.w, a2.w);

        const float4 pq = *(const float4*)(pp + (g0 + lr) * WD + col4);

        const float dil[4] = { max3(vL, v0, v1), max3(v0, v1, v2),
                               max3(v1, v2, v3), max3(v2, v3, vR) };
        const float ys[4]  = { a1.x, a1.y, a1.z, a1.w };
        const float ps[4]  = { pq.x, pq.y, pq.z, pq.w };

        #pragma unroll
        for (int j = 0; j < 4; ++j) {
            const float y     = ys[j];
            const float maskw = (dil[j] > 0.2f) ? 11.0f : 1.0f;

            // OMEGA=14, EPSILON=1, THETA=0.5, ALPHA=2.1, ratio = 0.5
            const float expo  = 2.1f - y;        // in (1.1, 2.1]
            const float pow_t = EXP2F(-expo);    // 0.5^expo in [0.233, 0.466]
            // A = 14*expo*2*pow_t/(1+pow_t)  (0.5^(expo-1) = 2*pow_t)
            const float Aa = __fdividef(28.0f * expo * pow_t, 1.0f + pow_t);
            const float Cc = fmaf(0.5f, Aa, -14.0f * log1p_fast(pow_t));

            const float d  = fabsf(ps[j] - y);
            // d^expo = exp2(expo*log2 d); d==0 -> exp2(-inf) = 0 (correct limit)
            const float pd    = EXP2F(expo * LOG2F(d));
            const float lossA = 14.0f * log1p_fast(pd);
            const float lossB = fmaf(Aa, d, -Cc);
            const float loss  = (d < 0.5f) ? lossA : lossB;

            lsum = fmaf(loss, maskw, lsum);
        }
    }

    // ---- block reduction (wave32 shuffle tree, deterministic) --------------
    #pragma unroll
    for (int off = 16; off > 0; off >>= 1)
        lsum += __shfl_down(lsum, off, 32);
    if ((tid & 31) == 0) wsum[tid >> 5] = lsum;
    __syncthreads();
    if (tid == 0) {
        float s = 0.0f;
        #pragma unroll
        for (int w = 0; w < THREADS / 32; ++w) s += wsum[w];
        partial[blk] = s;
    }
}

__global__ __launch_bounds__(THREADS)
void awing_reduce(const float* __restrict__ partial, float* __restrict__ out)
{
    __shared__ float wsum[THREADS / 32];
    float s = 0.0f;
    for (int i = threadIdx.x; i < NBLOCKS; i += THREADS)
        s += partial[i];
    #pragma unroll
    for (int off = 16; off > 0; off >>= 1)
        s += __shfl_down(s, off, 32);
    if ((threadIdx.x & 31) == 0) wsum[threadIdx.x >> 5] = s;
    __syncthreads();
    if (threadIdx.x == 0) {
        float t = 0.0f;
        #pragma unroll
        for (int w = 0; w < THREADS / 32; ++w) t += wsum[w];
        out[0] = t * (1.0f / (float)NTOT);       // mean
    }
}

extern "C" void kernel_launch(void* const* d_in, const int* in_sizes, int n_in,
                              void* d_out, int out_size, void* d_ws, size_t ws_size,
                              hipStream_t stream)
{
    const float* pred = (const float*)d_in[0];   // prediction
    const float* tgt  = (const float*)d_in[1];   // target
    float* out     = (float*)d_out;              // 1 element
    float* partial = (float*)d_ws;               // NBLOCKS floats (~35 KB)

    awing_main<<<NBLOCKS, THREADS, 0, stream>>>(pred, tgt, partial);
    awing_reduce<<<1, THREADS, 0, stream>>>(partial, out);
}